// ScorerV1_5205500362804
// MI455X (gfx1250) — compile-verified
//
#include <hip/hip_runtime.h>
#include <math.h>

// Problem constants (from reference)
#define B_      64
#define S_      128
#define H_      4096
#define NH_     32
#define HD_     128
#define MAXPOS_ 128

typedef float v2f __attribute__((ext_vector_type(2)));
typedef float v8f __attribute__((ext_vector_type(8)));

// ---------------------------------------------------------------------------
// Kernel 1: cond_sum[b,j] = sum_q condition[b,q,j]      (streams 128 MB)
// ---------------------------------------------------------------------------
__global__ __launch_bounds__(256) void colsum_ker(const float* __restrict__ x,
                                                  float* __restrict__ out) {
  int b = blockIdx.y;
  int j = blockIdx.x * 256 + threadIdx.x;
  const float* p = x + (size_t)b * S_ * H_ + j;
  float s = 0.f;
#pragma unroll 4
  for (int q = 0; q < S_; ++q) s += p[(size_t)q * H_];
  out[(size_t)b * H_ + j] = s;
}

// ---------------------------------------------------------------------------
// Kernel 2: pebias[k] = sum_q sum_h dist_emb[q-k+127, h]   (tiny)
// ---------------------------------------------------------------------------
__global__ __launch_bounds__(256) void pebias_ker(const float* __restrict__ demb,
                                                  float* __restrict__ pebias) {
  __shared__ float rowsum[2 * MAXPOS_ - 1];
  for (int d = threadIdx.x; d < 2 * MAXPOS_ - 1; d += 256) {
    float s = 0.f;
    for (int h = 0; h < NH_; ++h) s += demb[d * NH_ + h];
    rowsum[d] = s;
  }
  __syncthreads();
  int k = threadIdx.x;
  if (k < S_) {
    float s = 0.f;
    for (int q = 0; q < S_; ++q) s += rowsum[q - k + MAXPOS_ - 1];
    pebias[k] = s;
  }
}

// ---------------------------------------------------------------------------
// WMMA fp32 GEMM:  out[m,n] = sum_k A[m,k] * Bmat[k,n] (+ biasScale*bias[n])
//   K = N = 4096 compile-time. M = 64. One block (4 waves) per 16-wide
//   N strip; wave w owns k in [w*1024,(w+1)*1024) (in-block split-K),
//   partials combined through LDS in fixed order (deterministic).
//   K slice = 32 blocks of 32; scalar outer loop with compile-time trip
//   count, inner loop fully unrolled: 8 pipelined steps of (load next
//   fragments -> 4 WMMAs on current). Final block peeled so the hot loop
//   has no compares/selects and EXEC stays all-ones throughout.
//   WT=true : weight stored [N,K] (row n contiguous in k)  -> C̄ = c̄ @ W_w^T
//   WT=false: weight stored [K,N] (row k contiguous in n)  -> v  = C̄ @ Ws_w
// A-frag (16x4 f32, 2 VGPR): lane<16: M=lane, K={0,1}; lane>=16: K={2,3}
// B-frag (4x16  f32, 2 VGPR): lane half selects K pair, lane&15 = N
// D (16x16 f32, 8 VGPR): vgpr r -> M = r + 8*(lane>>4), N = lane&15
// ---------------------------------------------------------------------------
template <bool WT>
__global__ __launch_bounds__(128) void gemm64_wmma(const float* __restrict__ A,
                                                   const float* __restrict__ W,
                                                   const float* __restrict__ bias,
                                                   float biasScale,
                                                   float* __restrict__ out) {
  constexpr int KK     = H_;       // 4096
  constexpr int NN     = H_;       // 4096
  constexpr int KSLICE = KK / 4;   // 1024 per wave
  constexpr int NBLK   = KSLICE / 32;  // 32 blocks of 32 k

  __shared__ float red[4 * 1024];  // 4 waves x 64x16 tile = 16 KB

  const int tid  = threadIdx.x;
  const int w    = tid >> 5;       // wave id 0..3 (split-K slice)
  const int lane = tid & 31;
  const int lq   = lane & 15;
  const int hl   = lane >> 4;      // 0 or 1
  const int n    = blockIdx.x * 16 + lq;

  const int kBeg = w * KSLICE;

  // Per-lane base pointers at this wave's slice start (+ lane's K sub-offset)
  const float* A0 = A + (size_t)(0 * 16 + lq) * KK + kBeg + hl * 2;
  const float* A1 = A + (size_t)(1 * 16 + lq) * KK + kBeg + hl * 2;
  const float* A2 = A + (size_t)(2 * 16 + lq) * KK + kBeg + hl * 2;
  const float* A3 = A + (size_t)(3 * 16 + lq) * KK + kBeg + hl * 2;
  const float* Wb = WT ? (W + (size_t)n * KK + kBeg + hl * 2)
                       : (W + (size_t)(kBeg + hl * 2) * NN + n);

  v8f acc0 = {}, acc1 = {}, acc2 = {}, acc3 = {};

  // prime the pipeline: fragments for k-group 0 of this slice
  v2f bc;
  if (WT) bc = *(const v2f*)Wb;
  else { bc.x = Wb[0]; bc.y = Wb[NN]; }
  v2f ac0 = *(const v2f*)A0;
  v2f ac1 = *(const v2f*)A1;
  v2f ac2 = *(const v2f*)A2;
  v2f ac3 = *(const v2f*)A3;

  // ---- main: 31 blocks of 32 k (compile-time trip count, uniform) ----------
  for (int blk = 0; blk < NBLK - 1; ++blk) {
    const int base = blk * 32;     // offset within slice
    {  // one prefetch per block, ~512 floats ahead, clamped inside slice
      int kp = base + 512; if (kp > KSLICE - 32) kp = KSLICE - 32;
      if (WT) __builtin_prefetch(Wb + kp, 0, 1);
      else    __builtin_prefetch(Wb + (size_t)kp * NN, 0, 1);
    }
#pragma unroll
    for (int kk = 4; kk <= 32; kk += 4) {
      // load next step's fragments (overlap with WMMAs below)
      v2f bn;
      if (WT) bn = *(const v2f*)(Wb + base + kk);
      else { bn.x = Wb[(size_t)(base + kk) * NN];
             bn.y = Wb[(size_t)(base + kk + 1) * NN]; }
      v2f an0 = *(const v2f*)(A0 + base + kk);
      v2f an1 = *(const v2f*)(A1 + base + kk);
      v2f an2 = *(const v2f*)(A2 + base + kk);
      v2f an3 = *(const v2f*)(A3 + base + kk);

      acc0 = __builtin_amdgcn_wmma_f32_16x16x4_f32(false, ac0, false, bc,
                                                   (short)0, acc0, false, false);
      acc1 = __builtin_amdgcn_wmma_f32_16x16x4_f32(false, ac1, false, bc,
                                                   (short)0, acc1, false, false);
      acc2 = __builtin_amdgcn_wmma_f32_16x16x4_f32(false, ac2, false, bc,
                                                   (short)0, acc2, false, false);
      acc3 = __builtin_amdgcn_wmma_f32_16x16x4_f32(false, ac3, false, bc,
                                                   (short)0, acc3, false, false);
      bc = bn; ac0 = an0; ac1 = an1; ac2 = an2; ac3 = an3;
    }
  }

  // ---- peeled final block: last step has no next-load (kk<32 is constexpr) --
  {
    const int base = (NBLK - 1) * 32;
#pragma unroll
    for (int kk = 4; kk <= 32; kk += 4) {
      v2f bn, an0, an1, an2, an3;
      if (kk < 32) {
        if (WT) bn = *(const v2f*)(Wb + base + kk);
        else { bn.x = Wb[(size_t)(base + kk) * NN];
               bn.y = Wb[(size_t)(base + kk + 1) * NN]; }
        an0 = *(const v2f*)(A0 + base + kk);
        an1 = *(const v2f*)(A1 + base + kk);
        an2 = *(const v2f*)(A2 + base + kk);
        an3 = *(const v2f*)(A3 + base + kk);
      }
      acc0 = __builtin_amdgcn_wmma_f32_16x16x4_f32(false, ac0, false, bc,
                                                   (short)0, acc0, false, false);
      acc1 = __builtin_amdgcn_wmma_f32_16x16x4_f32(false, ac1, false, bc,
                                                   (short)0, acc1, false, false);
      acc2 = __builtin_amdgcn_wmma_f32_16x16x4_f32(false, ac2, false, bc,
                                                   (short)0, acc2, false, false);
      acc3 = __builtin_amdgcn_wmma_f32_16x16x4_f32(false, ac3, false, bc,
                                                   (short)0, acc3, false, false);
      if (kk < 32) { bc = bn; ac0 = an0; ac1 = an1; ac2 = an2; ac3 = an3; }
    }
  }

  // --- combine split-K partials through LDS (fixed order => deterministic) --
  float* my = red + w * 1024;
#pragma unroll
  for (int r = 0; r < 8; ++r) {
    my[(0 * 8 + r) * 32 + lane] = acc0[r];
    my[(1 * 8 + r) * 32 + lane] = acc1[r];
    my[(2 * 8 + r) * 32 + lane] = acc2[r];
    my[(3 * 8 + r) * 32 + lane] = acc3[r];
  }
  __syncthreads();

#pragma unroll
  for (int e = tid; e < 1024; e += 128) {
    float s = (red[e] + red[1024 + e]) + (red[2048 + e] + red[3072 + e]);
    const int lane_e = e & 31;
    const int rr     = (e >> 5) & 7;
    const int t      = e >> 8;
    const int m      = t * 16 + rr + ((lane_e >> 4) << 3);
    const int ng     = blockIdx.x * 16 + (lane_e & 15);
    const float bv   = bias ? biasScale * bias[ng] : 0.0f;
    out[(size_t)m * NN + ng] = s + bv;
  }
}

// ---------------------------------------------------------------------------
// Kernel 4: dotWb[b] = sum_j Cbar[b,j] * Ws_b[j]         (tiny)
// ---------------------------------------------------------------------------
__global__ __launch_bounds__(256) void dotwb_ker(const float* __restrict__ Cbar,
                                                 const float* __restrict__ wsb,
                                                 float* __restrict__ dotWb) {
  __shared__ float red[256];
  int b = blockIdx.x;
  float s = 0.f;
  for (int j = threadIdx.x; j < H_; j += 256)
    s += Cbar[(size_t)b * H_ + j] * wsb[j];
  red[threadIdx.x] = s;
  __syncthreads();
  for (int off = 128; off > 0; off >>= 1) {
    if (threadIdx.x < off) red[threadIdx.x] += red[threadIdx.x + off];
    __syncthreads();
  }
  if (threadIdx.x == 0) dotWb[b] = red[0];
}

// ---------------------------------------------------------------------------
// Kernel 5: raw[b,k] = v[b,:] . X_norm[b,k,:]            (streams 128 MB)
// ---------------------------------------------------------------------------
__global__ __launch_bounds__(256) void rowdot_ker(const float* __restrict__ X,
                                                  const float* __restrict__ vv,
                                                  float* __restrict__ raw) {
  __shared__ float red[256];
  const int bk = blockIdx.x;       // b*S + k
  const int b  = bk >> 7;          // /128
  const float4* xp = (const float4*)(X + (size_t)bk * H_);
  const float4* vp = (const float4*)(vv + (size_t)b * H_);
  float s = 0.f;
#pragma unroll
  for (int m = threadIdx.x; m < H_ / 4; m += 256) {
    float4 a = xp[m], c = vp[m];
    s += a.x * c.x + a.y * c.y + a.z * c.z + a.w * c.w;
  }
  red[threadIdx.x] = s;
  __syncthreads();
  for (int off = 128; off > 0; off >>= 1) {
    if (threadIdx.x < off) red[threadIdx.x] += red[threadIdx.x + off];
    __syncthreads();
  }
  if (threadIdx.x == 0) raw[bk] = red[0];
}

// ---------------------------------------------------------------------------
// Kernel 6: masked softmax over k of (raw + dotWb[b] + pebias[k]) * INV
// ---------------------------------------------------------------------------
__global__ __launch_bounds__(S_) void softmax_ker(const float* __restrict__ raw,
                                                  const float* __restrict__ dotWb,
                                                  const float* __restrict__ pebias,
                                                  const int* __restrict__ mask,
                                                  float* __restrict__ out) {
  __shared__ float red[S_];
  const int b = blockIdx.x, k = threadIdx.x;
  // 1/(NH * S * sqrt(HD))
  const float INV = 1.0f / (NH_ * (float)S_ * 11.313708498984760f);
  float val = (raw[b * S_ + k] + dotWb[b] + pebias[k]) * INV;
  if (mask[b * S_ + k] == 0) val = -INFINITY;
  red[k] = val;
  __syncthreads();
  for (int off = 64; off > 0; off >>= 1) {
    if (k < off) red[k] = fmaxf(red[k], red[k + off]);
    __syncthreads();
  }
  const float mx = red[0];
  __syncthreads();
  const float e = __expf(val - mx);
  red[k] = e;
  __syncthreads();
  for (int off = 64; off > 0; off >>= 1) {
    if (k < off) red[k] += red[k + off];
    __syncthreads();
  }
  out[b * S_ + k] = e / red[0];
}

// ---------------------------------------------------------------------------
extern "C" void kernel_launch(void* const* d_in, const int* in_sizes, int n_in,
                              void* d_out, int out_size, void* d_ws, size_t ws_size,
                              hipStream_t stream) {
  const float* X_norm    = (const float*)d_in[0];   // [B,S,H]
  const int*   mask      = (const int*)  d_in[1];   // [B,S]
  const float* condition = (const float*)d_in[2];   // [B,S,H]
  const float* W_w       = (const float*)d_in[3];   // [H,H] row-major [i,j]
  const float* W_b       = (const float*)d_in[4];   // [H]
  const float* Ws_w      = (const float*)d_in[5];   // [H,H]
  const float* Ws_b      = (const float*)d_in[6];   // [H]
  const float* dist_emb  = (const float*)d_in[7];   // [255,NH]
  float* out = (float*)d_out;                       // [B,S]

  float* ws       = (float*)d_ws;
  float* cond_sum = ws;                       // B*H
  float* Cbar     = cond_sum + B_ * H_;       // B*H
  float* vvec     = Cbar     + B_ * H_;       // B*H
  float* raw      = vvec     + B_ * H_;       // B*S
  float* dotWb    = raw      + B_ * S_;       // B
  float* pebias   = dotWb    + B_;            // S

  // 1) sum condition over queries
  colsum_ker<<<dim3(H_ / 256, B_), 256, 0, stream>>>(condition, cond_sum);
  // 2) relative-position bias (independent of inputs above)
  pebias_ker<<<1, 256, 0, stream>>>(dist_emb, pebias);
  // 3) Cbar = cond_sum @ W_w^T + S*W_b       (weight indexed [n*K+k])
  gemm64_wmma<true ><<<H_ / 16, 128, 0, stream>>>(cond_sum, W_w, W_b,
                                                  (float)S_, Cbar);
  // 4) v = Cbar @ Ws_w                       (weight indexed [k*N+n])
  gemm64_wmma<false><<<H_ / 16, 128, 0, stream>>>(Cbar, Ws_w, nullptr,
                                                  0.0f, vvec);
  // 5) dotWb[b] = Cbar[b,:] . Ws_b
  dotwb_ker<<<B_, 256, 0, stream>>>(Cbar, Ws_b, dotWb);
  // 6) raw[b,k] = v[b,:] . X_norm[b,k,:]
  rowdot_ker<<<B_ * S_, 256, 0, stream>>>(X_norm, vvec, raw);
  // 7) masked softmax
  softmax_ker<<<B_, S_, 0, stream>>>(raw, dotWb, pebias, mask, out);
}